// NNConv1Layer_concatmsg_61632780698137
// MI455X (gfx1250) — compile-verified
//
#include <hip/hip_runtime.h>
#include <hip/hip_bf16.h>

// ---------------------------------------------------------------------------
// Hetero-graph NNConv, fused per-edge pass using V_WMMA_F32_16X16X4_F32 for
// the theta = ea @ nnW.T + nnb GEMM (dominant FLOP term), f32 HW atomics for
// segment_sum, scalar per-node epilogue. f32 end-to-end (reference dtype).
// Memory-bound: ~430 MB traffic, ~4.5 GFLOP -> optimize instruction overhead
// around the fixed traffic. Edge kernel templated on (FS, DE) so NT/KC are
// compile-time and the WMMA tile loops fully unroll.
// ---------------------------------------------------------------------------

typedef float v2f __attribute__((ext_vector_type(2)));
typedef float v8f __attribute__((ext_vector_type(8)));

#define OUTD 5          // OUT
#define COLS_MAX 80     // OUT * max(Fs)
#define TH_STRIDE 85    // 80 padded -> conflict-free LDS (85 % 64 = 21)
#define WPB 4           // waves per block
#define EPB (WPB * 32)  // edges per block

// Each wave owns 32 edges; 4 waves share one staged nnW.T copy.
// theta tile (32 edges x cols) via WMMA f32 16x16x4 (bias folded into C),
// staged to LDS, then per-lane finish: gather x_src row, 5-dot, 5 atomics.
template <int FS, int DE>
__global__ __launch_bounds__(EPB) void nnconv_edge_wmma(
    const float* __restrict__ x_src,
    const int*   __restrict__ ei,      // [2*E]: src rows then dst rows
    const float* __restrict__ ea,
    const float* __restrict__ nnW,     // (5*FS, DE) row-major
    const float* __restrict__ nnb,     // (5*FS)
    float* __restrict__ agg,           // (N_dst, 35)
    int colbase, int E)
{
  constexpr int cols = OUTD * FS;          // 80 or 40
  constexpr int NT   = (cols + 15) >> 4;   // 5 or 3
  constexpr int KC   = (DE + 3) >> 2;      // 2 (de=8) or 1 (de=1)

  const int tid  = threadIdx.x;
  const int wv   = tid >> 5;               // wave id 0..3
  const int lane = tid & 31;
  const int base = blockIdx.x * EPB + wv * 32;

  __shared__ float Wt_s[8 * COLS_MAX];         // nnW.T zero-padded to 8 x 80
  __shared__ float ea_s[EPB * 8];              // edge attrs padded to 8
  __shared__ float th_s[EPB * TH_STRIDE];      // theta tiles (per wave slice)

  // ---- stage weights (shared) + edge attrs (per lane) into LDS
  for (int i = tid; i < 8 * COLS_MAX; i += EPB) Wt_s[i] = 0.0f;
#pragma unroll
  for (int d = 0; d < 8; ++d) ea_s[tid * 8 + d] = 0.0f;
  __syncthreads();
  for (int i = tid; i < cols * DE; i += EPB) {
    const int col = i / DE;
    const int d   = i - col * DE;
    Wt_s[d * COLS_MAX + col] = nnW[i];         // Wt[d][col] = nnW[col][d]
  }
  const int e = base + lane;
  if (e < E) {
    const float* er = ea + (long)e * DE;
#pragma unroll
    for (int d = 0; d < DE; ++d) ea_s[tid * 8 + d] = er[d];
  }
  __syncthreads();

  // ---- WMMA: theta tiles, K accumulated in 4-chunks.
  // A 16x4 f32: VGPR0 = K0 (lanes 0-15) / K2 (lanes 16-31), VGPR1 = K1/K3.
  // B 4x16 (mirrors C): VGPR0 = K-row 0 / 2, VGPR1 = K-row 1 / 3.
  // C/D 16x16: VGPR r = row r (lanes 0-15) / row r+8 (lanes 16-31).
  const int half = lane >> 4;
  const int l16  = lane & 15;

#pragma unroll
  for (int mt = 0; mt < 2; ++mt) {             // two 16-edge M tiles per wave
    const int arow = wv * 32 + mt * 16 + l16;
#pragma unroll
    for (int nt = 0; nt < NT; ++nt) {
      const int n0 = nt * 16;
      const float cb = (n0 + l16 < cols) ? nnb[n0 + l16] : 0.0f;  // bias in C
      v8f c = {cb, cb, cb, cb, cb, cb, cb, cb};
#pragma unroll
      for (int kc = 0; kc < KC; ++kc) {
        const int k0 = kc * 4 + half * 2;
        v2f a, b;
        a.x = ea_s[arow * 8 + k0];
        a.y = ea_s[arow * 8 + k0 + 1];
        b.x = Wt_s[(k0    ) * COLS_MAX + n0 + l16];
        b.y = Wt_s[(k0 + 1) * COLS_MAX + n0 + l16];
        c = __builtin_amdgcn_wmma_f32_16x16x4_f32(
            /*neg_a=*/false, a, /*neg_b=*/false, b,
            /*c_mod=*/(short)0, c, /*reuse_a=*/false, /*reuse_b=*/false);
      }
#pragma unroll
      for (int r = 0; r < 8; ++r) {            // D -> LDS (conflict-free)
        th_s[(wv * 32 + mt * 16 + half * 8 + r) * TH_STRIDE + n0 + l16] = c[r];
      }
    }
  }
  __syncthreads();

  // ---- per-edge finish: msg[o] = sum_f x[f] * theta[e][f*5+o]; scatter-add
  if (e < E) {
    const int snode = ei[e];
    const int dnode = ei[E + e];
    const float4* xp = (const float4*)(x_src + (long)snode * FS);
    float xv[FS];
#pragma unroll
    for (int q = 0; q < FS / 4; ++q) {
      const float4 t = xp[q];
      xv[4 * q + 0] = t.x; xv[4 * q + 1] = t.y;
      xv[4 * q + 2] = t.z; xv[4 * q + 3] = t.w;
    }
    float acc[OUTD] = {0.f, 0.f, 0.f, 0.f, 0.f};
    const float* th = &th_s[tid * TH_STRIDE];
#pragma unroll
    for (int f = 0; f < FS; ++f) {
      const float xvf = xv[f];
#pragma unroll
      for (int o = 0; o < OUTD; ++o) acc[o] += xvf * th[f * OUTD + o];
    }
    float* dp = agg + (long)dnode * 35 + colbase;
#pragma unroll
    for (int o = 0; o < OUTD; ++o) unsafeAtomicAdd(dp + o, acc[o]);
  }
}

struct FinishParams {
  const float* rootW[7];   // each (5, 16) row-major
  const float* bias[7];    // each (5)
  const float* linW;       // (35)
  const float* linb;       // (1)
};

// per-node epilogue: z = sigmoid(agg + x@rootW.T + bias); out = sigmoid(z@linW + b)
__global__ void node_finish_kernel(const float* __restrict__ x, int N,
                                   const float* __restrict__ agg,
                                   FinishParams p, float* __restrict__ out)
{
  const int n = blockIdx.x * blockDim.x + threadIdx.x;
  if (n >= N) return;
  float xv[16];
  {
    const float4* xp = (const float4*)(x + (long)n * 16);
    const float4 t0 = xp[0], t1 = xp[1], t2 = xp[2], t3 = xp[3];
    xv[0]=t0.x;  xv[1]=t0.y;  xv[2]=t0.z;  xv[3]=t0.w;
    xv[4]=t1.x;  xv[5]=t1.y;  xv[6]=t1.z;  xv[7]=t1.w;
    xv[8]=t2.x;  xv[9]=t2.y;  xv[10]=t2.z; xv[11]=t2.w;
    xv[12]=t3.x; xv[13]=t3.y; xv[14]=t3.z; xv[15]=t3.w;
  }
  const float* ag = agg + (long)n * 35;
  float s = 0.0f;
  for (int j = 0; j < 7; ++j) {
    const float* rw = p.rootW[j];
    const float* bs = p.bias[j];
#pragma unroll
    for (int o = 0; o < OUTD; ++o) {
      float v = ag[j * OUTD + o] + bs[o];
      const float* w = rw + o * 16;
#pragma unroll
      for (int f = 0; f < 16; ++f) v += xv[f] * w[f];
      const float z = 1.0f / (1.0f + __expf(-v));
      s += z * p.linW[j * OUTD + o];
    }
  }
  s += p.linb[0];
  out[n] = 1.0f / (1.0f + __expf(-s));
}

__global__ void zero_f32_kernel(float* __restrict__ p, long n) {
  const long i = (long)blockIdx.x * blockDim.x + threadIdx.x;
  if (i < n) p[i] = 0.0f;
}

// ---------------------------------------------------------------------------
// Input layout (setup_inputs() insertion order):
//   0..2   : x_ind (100000x16), x_org (30000x16), x_ext (10000x8)
//   3+2r   : ei_<rel r> (2 x E int32)      r = 0..13 in RELS order
//   4+2r   : ea_<rel r> (E x de f32)
//   31+4r  : params[rel r]: nnW, nnb, rootW, bias   (dict insertion order)
//   87..88 : lin_ind W (35), b (1);  89..90 : lin_org W, b
// Output: out_ind (100000) then out_org (30000), f32.
// ---------------------------------------------------------------------------
extern "C" void kernel_launch(void* const* d_in, const int* in_sizes, int n_in,
                              void* d_out, int out_size, void* d_ws, size_t ws_size,
                              hipStream_t stream) {
  const float* xs[3] = {(const float*)d_in[0], (const float*)d_in[1], (const float*)d_in[2]};
  const int fs[3] = {16, 16, 8};
  const int N_ind = in_sizes[0] / 16;
  const int N_org = in_sizes[1] / 16;

  float* agg_ind = (float*)d_ws;
  float* agg_org = agg_ind + (size_t)N_ind * 35;

  const long ztot = (long)(N_ind + N_org) * 35;
  zero_f32_kernel<<<(int)((ztot + 255) / 256), 256, 0, stream>>>(agg_ind, ztot);

  // RELS order: {src_type, dst(0=ind,1=org), de, slot}
  struct RelDesc { int src, dst, de, slot; };
  const RelDesc rels[14] = {
    {0,0,8,0},{1,0,8,1},{2,0,8,2},{0,0,8,3},{1,0,8,4},{2,0,8,5},{1,0,1,6},
    {0,1,8,0},{1,1,8,1},{2,1,8,2},{0,1,1,3},{0,1,8,4},{1,1,8,5},{2,1,8,6}
  };

  for (int r = 0; r < 14; ++r) {
    const int*   ei  = (const int*)d_in[3 + 2 * r];
    const float* ea  = (const float*)d_in[4 + 2 * r];
    const int    E   = in_sizes[3 + 2 * r] / 2;
    const int    pb  = 31 + 4 * r;
    const float* nnW = (const float*)d_in[pb + 0];
    const float* nnb = (const float*)d_in[pb + 1];
    float* agg = (rels[r].dst == 0) ? agg_ind : agg_org;
    const int F  = fs[rels[r].src];
    const int de = rels[r].de;
    const int blocks = (E + EPB - 1) / EPB;
    const float* x = xs[rels[r].src];
    const int cb = rels[r].slot * OUTD;
    if (F == 16 && de == 8) {
      nnconv_edge_wmma<16, 8><<<blocks, EPB, 0, stream>>>(x, ei, ea, nnW, nnb, agg, cb, E);
    } else if (F == 8) {
      nnconv_edge_wmma<8, 8><<<blocks, EPB, 0, stream>>>(x, ei, ea, nnW, nnb, agg, cb, E);
    } else {
      nnconv_edge_wmma<16, 1><<<blocks, EPB, 0, stream>>>(x, ei, ea, nnW, nnb, agg, cb, E);
    }
  }

  FinishParams pind, porg;
  for (int j = 0; j < 7; ++j) {
    pind.rootW[j] = (const float*)d_in[31 + 4 * j + 2];
    pind.bias[j]  = (const float*)d_in[31 + 4 * j + 3];
    porg.rootW[j] = (const float*)d_in[31 + 4 * (7 + j) + 2];
    porg.bias[j]  = (const float*)d_in[31 + 4 * (7 + j) + 3];
  }
  pind.linW = (const float*)d_in[87];
  pind.linb = (const float*)d_in[88];
  porg.linW = (const float*)d_in[89];
  porg.linb = (const float*)d_in[90];

  float* out = (float*)d_out;
  node_finish_kernel<<<(N_ind + 255) / 256, 256, 0, stream>>>(xs[0], N_ind, agg_ind, pind, out);
  node_finish_kernel<<<(N_org + 255) / 256, 256, 0, stream>>>(xs[1], N_org, agg_org, porg, out + N_ind);
}